// LiftSplat_50087908606309
// MI455X (gfx1250) — compile-verified
//
#include <hip/hip_runtime.h>
#include <cstdint>
#include <cstddef>

// ---------------- problem constants (from reference) ----------------
#define B_      4
#define NX_     400
#define NY_     200
#define C_      256
#define NPRIME_ 173184              // B * N_CAMS * D * FH * FW
#define NXY_    (NX_ * NY_)         // 80000
#define NVOX_   (B_ * NXY_)         // 320000 voxels (NZ == 1, gz is always 0)

typedef float v4f __attribute__((ext_vector_type(4)));
typedef int   v4i __attribute__((ext_vector_type(4)));

#define AS1 __attribute__((address_space(1)))
#define AS3 __attribute__((address_space(3)))

// ---------------- CDNA5 feature probes ----------------
#if defined(__has_builtin)
#  if __has_builtin(__builtin_amdgcn_global_load_async_to_lds_b32)
#    define HAVE_ASYNC_LDS 1
#  endif
#endif
#ifndef HAVE_ASYNC_LDS
#  define HAVE_ASYNC_LDS 0
#endif

static __device__ __forceinline__ void wait_asynccnt0() {
#if defined(__has_builtin) && __has_builtin(__builtin_amdgcn_s_wait_asynccnt)
    __builtin_amdgcn_s_wait_asynccnt(0);
#else
    asm volatile("s_wait_asynccnt 0" ::: "memory");
#endif
}

static __device__ __forceinline__ void atomic_add_f32(float* p, float v) {
#if defined(__has_builtin) && __has_builtin(__builtin_amdgcn_global_atomic_fadd_f32)
    __builtin_amdgcn_global_atomic_fadd_f32((AS1 float*)p, v);
#else
    (void)__hip_atomic_fetch_add(p, v, __ATOMIC_RELAXED, __HIP_MEMORY_SCOPE_AGENT);
#endif
}

// ---------------- scatter-add ----------------
// One 64-thread group per point; each thread handles 4 consecutive channels
// (float4 load of x). VOXMAJOR=true scatters into voxel-major scratch
// [b][x][y][c] so a point's 256 atomics hit 8 contiguous cachelines.
// VOXMAJOR=false scatters straight into the final [b][c][x][y] layout.
template <bool VOXMAJOR>
__global__ __launch_bounds__(256)
void scatter_kernel(const float* __restrict__ x,
                    const int*   __restrict__ geom,
                    float*       __restrict__ out) {
    const unsigned gid = blockIdx.x * 256u + threadIdx.x;
    const unsigned p   = gid >> 6;            // point index (wave-uniform)
    const unsigned c0  = (gid & 63u) << 2;    // first of 4 channels
    if (p >= NPRIME_) return;

    const v4i g  = ((const v4i*)geom)[p];     // {gx, gy, gz, gb}; gz == 0
    const v4f xv = __builtin_nontemporal_load(
        (const v4f*)(x + (size_t)p * C_) + (c0 >> 2));

    const unsigned gx = (unsigned)g.x, gy = (unsigned)g.y, gb = (unsigned)g.w;

    if (VOXMAJOR) {
        // [b][x][y][c]: 4 contiguous atomics per thread, wave covers 128
        // contiguous floats (one 512B burst into a 1KB voxel slab)
        const size_t base = ((size_t)((gb * NX_ + gx) * NY_ + gy)) * C_ + c0;
        float* o = out + base;
        atomic_add_f32(o + 0, xv.x);
        atomic_add_f32(o + 1, xv.y);
        atomic_add_f32(o + 2, xv.z);
        atomic_add_f32(o + 3, xv.w);
    } else {
        // [b][c][x][y]: channel stride = NXY (fallback when ws too small)
        const size_t base = ((size_t)(gb * C_ + c0)) * NXY_
                          + (size_t)gx * NY_ + gy;
        atomic_add_f32(out + base + 0 * (size_t)NXY_, xv.x);
        atomic_add_f32(out + base + 1 * (size_t)NXY_, xv.y);
        atomic_add_f32(out + base + 2 * (size_t)NXY_, xv.z);
        atomic_add_f32(out + base + 3 * (size_t)NXY_, xv.w);
    }
}

// ---------------- transpose [NXY][C] -> [C][NXY] per batch ----------------
// 32x32 tiles through LDS. Global->LDS uses the gfx1250 async-to-LDS path
// (ASYNCcnt + s_wait_asynccnt) when available; reads of the scratch are
// coalesced along C, writes of the output are coalesced along NXY and
// nontemporal (each output element written exactly once).
__global__ __launch_bounds__(256)
void transpose_kernel(const float* __restrict__ in, float* __restrict__ out) {
    __shared__ float tile[32][33];            // +1 pad: conflict-free column reads

    const unsigned b  = blockIdx.z;
    const unsigned r0 = blockIdx.x * 32u;     // spatial (NXY) tile origin
    const unsigned c0 = blockIdx.y * 32u;     // channel tile origin
    const float* inb  = in  + (size_t)b * NXY_ * C_;
    float*       outb = out + (size_t)b * C_ * NXY_;

    const unsigned tx = threadIdx.x;          // 0..31
    const unsigned ty = threadIdx.y;          // 0..7

#if HAVE_ASYNC_LDS
    #pragma unroll
    for (int i = 0; i < 4; ++i) {
        const unsigned r = ty + 8u * i;
        const float* src = inb + (size_t)(r0 + r) * C_ + (c0 + tx);
        __builtin_amdgcn_global_load_async_to_lds_b32(
            (AS1 int*)src, (AS3 int*)&tile[r][tx], 0, 0);
    }
    wait_asynccnt0();
#else
    #pragma unroll
    for (int i = 0; i < 4; ++i) {
        const unsigned r = ty + 8u * i;
        tile[r][tx] = inb[(size_t)(r0 + r) * C_ + (c0 + tx)];
    }
#endif
    __syncthreads();

    #pragma unroll
    for (int i = 0; i < 4; ++i) {
        const unsigned c = ty + 8u * i;
        __builtin_nontemporal_store(
            tile[tx][c], outb + (size_t)(c0 + c) * NXY_ + (r0 + tx));
    }
}

// ---------------- launch ----------------
extern "C" void kernel_launch(void* const* d_in, const int* in_sizes, int n_in,
                              void* d_out, int out_size, void* d_ws, size_t ws_size,
                              hipStream_t stream) {
    (void)in_sizes; (void)n_in;
    const float* x    = (const float*)d_in[0];   // (NPRIME, C) f32
    const int*   geom = (const int*)d_in[1];     // (NPRIME, 4) i32
    float*       out  = (float*)d_out;           // (B, C, NX, NY) f32

    const size_t vox_bytes      = (size_t)NVOX_ * C_ * sizeof(float); // 327.68 MB
    const int    scatter_blocks = (NPRIME_ * 64) / 256;               // 43296 exact

    if (ws_size >= vox_bytes) {
        // Staged: voxel-major accumulate (line-local atomics), then transpose.
        float* ws = (float*)d_ws;
        (void)hipMemsetAsync(ws, 0, vox_bytes, stream);
        scatter_kernel<true><<<scatter_blocks, 256, 0, stream>>>(x, geom, ws);
        dim3 tgrid(NXY_ / 32, C_ / 32, B_);      // (2500, 8, 4), exact tiling
        dim3 tblk(32, 8, 1);
        transpose_kernel<<<tgrid, tblk, 0, stream>>>(ws, out);
    } else {
        // Fallback: zero output and scatter directly into final layout.
        (void)hipMemsetAsync(out, 0, (size_t)out_size * sizeof(float), stream);
        scatter_kernel<false><<<scatter_blocks, 256, 0, stream>>>(x, geom, out);
    }
}